// SphericalExpansion_21174188769328
// MI455X (gfx1250) — compile-verified
//
#include <hip/hip_runtime.h>
#include <hip/hip_bf16.h>

typedef __attribute__((ext_vector_type(2))) float v2f;
typedef __attribute__((ext_vector_type(8))) float v8f;

#define PI_OVER_RCUT 0.6283185307179586f   // pi / 5.0
#define NORM_1_SQRT32 0.17677669529663689f // 1/sqrt(32)

// ---------------------------------------------------------------------------
// Per-edge basis: 12 radial terms (Chebyshev sin recurrence) + 16 real SH.
// ---------------------------------------------------------------------------
__device__ __forceinline__ void edge_basis(float vx, float vy, float vz,
                                           float phi[12], float sh[16]) {
    float r2    = vx * vx + vy * vy + vz * vz + 1e-12f;
    float r     = sqrtf(r2);
    float inv_r = 1.0f / r;
    float x = vx * inv_r, y = vy * inv_r, z = vz * inv_r;

    float a = PI_OVER_RCUT * r;
    float s1, c1;
    __sincosf(a, &s1, &c1);
    float twoc  = 2.0f * c1;
    float scale = inv_r * NORM_1_SQRT32;
    float sm1 = 0.0f, s = s1;
    #pragma unroll
    for (int n = 0; n < 12; ++n) {
        phi[n] = s * scale;
        float nxt = twoc * s - sm1;
        sm1 = s;
        s   = nxt;
    }

    float x2 = x * x, y2 = y * y, z2 = z * z;
    sh[0]  = 0.28209479177387814f;
    sh[1]  = 0.4886025119029199f * y;
    sh[2]  = 0.4886025119029199f * z;
    sh[3]  = 0.4886025119029199f * x;
    sh[4]  = 1.0925484305920792f * x * y;
    sh[5]  = 1.0925484305920792f * y * z;
    sh[6]  = 0.31539156525252005f * (3.0f * z2 - 1.0f);
    sh[7]  = 1.0925484305920792f * x * z;
    sh[8]  = 0.5462742152960396f * (x2 - y2);
    sh[9]  = 0.5900435899266435f * y * (3.0f * x2 - y2);
    sh[10] = 2.890611442640554f * x * y * z;
    sh[11] = 0.4570457994644658f * y * (5.0f * z2 - 1.0f);
    sh[12] = 0.3731763325901154f * (5.0f * z2 * z - 3.0f * z);
    sh[13] = 0.4570457994644658f * x * (5.0f * z2 - 1.0f);
    sh[14] = 1.445305721320277f * z * (x2 - y2);
    sh[15] = 0.5900435899266435f * x * (x2 - y2);
}

// B tiles of W (12x32), W[k][l*8+n] = rw[l*96 + k*8 + n]; WMMA B layout.
__device__ __forceinline__ void load_B(const float* __restrict__ rw, int lane,
                                       v2f b[2][3]) {
    const int m = lane & 15, half = lane >> 4;
    #pragma unroll
    for (int u = 0; u < 2; ++u)
        #pragma unroll
        for (int ks = 0; ks < 3; ++ks) {
            int col = 16 * u + m;
            int l = col >> 3, n = col & 7;
            b[u][ks][0] = rw[l * 96 + (4 * ks + 2 * half + 0) * 8 + n];
            b[u][ks][1] = rw[l * 96 + (4 * ks + 2 * half + 1) * 8 + n];
        }
}

// rad(32 edges x 32 cols) = phi @ W via V_WMMA_F32_16X16X4_F32, K=12 (3 steps).
// buf = per-wave 32x32 LDS scratch. Full-EXEC required: call unconditionally.
__device__ __forceinline__ void wmma_rad(float (*buf)[32], int lane,
                                         const v2f b[2][3], const float phi[12],
                                         float rad[32]) {
    #pragma unroll
    for (int k = 0; k < 12; ++k) buf[lane][k] = phi[k];
    __syncthreads();

    const int m = lane & 15, half = lane >> 4;
    v2f a[2][3];
    #pragma unroll
    for (int t = 0; t < 2; ++t)
        #pragma unroll
        for (int ks = 0; ks < 3; ++ks) {
            a[t][ks][0] = buf[16 * t + m][4 * ks + 2 * half + 0];
            a[t][ks][1] = buf[16 * t + m][4 * ks + 2 * half + 1];
        }
    __syncthreads();

    v8f acc[2][2];
    #pragma unroll
    for (int t = 0; t < 2; ++t)
        #pragma unroll
        for (int u = 0; u < 2; ++u) {
            v8f c = {0.f, 0.f, 0.f, 0.f, 0.f, 0.f, 0.f, 0.f};
            #pragma unroll
            for (int ks = 0; ks < 3; ++ks)
                c = __builtin_amdgcn_wmma_f32_16x16x4_f32(
                        false, a[t][ks], false, b[u][ks],
                        (short)0, c, false, false);
            acc[t][u] = c;
        }

    // C layout: VGPR r holds M = r + 8*half, N = lane%16 (per 16-col tile).
    #pragma unroll
    for (int t = 0; t < 2; ++t)
        #pragma unroll
        for (int u = 0; u < 2; ++u)
            #pragma unroll
            for (int rr = 0; rr < 8; ++rr)
                buf[16 * t + rr + 8 * half][16 * u + m] = acc[t][u][rr];
    __syncthreads();

    #pragma unroll
    for (int i = 0; i < 8; ++i) {
        float4 q = *reinterpret_cast<const float4*>(&buf[lane][4 * i]);
        rad[4 * i + 0] = q.x; rad[4 * i + 1] = q.y;
        rad[4 * i + 2] = q.z; rad[4 * i + 3] = q.w;
    }
    __syncthreads();  // protect buf reuse (next chunk's phi)
}

// ---------------------------------------------------------------------------
// Sorted-path kernels
// ---------------------------------------------------------------------------
__global__ void fill_u32_kernel(unsigned* __restrict__ p, unsigned v, int n) {
    int i = blockIdx.x * blockDim.x + threadIdx.x;
    if (i < n) p[i] = v;
}

__global__ void hist_kernel(const int* __restrict__ c, const int* __restrict__ s,
                            unsigned* __restrict__ count, int n) {
    int e = blockIdx.x * 256 + threadIdx.x;
    if (e < n) atomicAdd(&count[c[e] * 4 + s[e]], 1u);
}

// Per-block exclusive scan of 256 u32 (Hillis-Steele), block totals to bsums.
__global__ void scan_block_kernel(const unsigned* __restrict__ in,
                                  unsigned* __restrict__ out,
                                  unsigned* __restrict__ bsums, int n) {
    __shared__ unsigned tmp[256];
    const int t = threadIdx.x;
    const int g = blockIdx.x * 256 + t;
    unsigned v = (g < n) ? in[g] : 0u;
    tmp[t] = v;
    __syncthreads();
    #pragma unroll
    for (int d = 1; d < 256; d <<= 1) {
        unsigned add = (t >= d) ? tmp[t - d] : 0u;
        __syncthreads();
        tmp[t] += add;
        __syncthreads();
    }
    if (g < n) out[g] = tmp[t] - v;        // exclusive
    if (t == 255) bsums[blockIdx.x] = tmp[255];
}

// Single-block exclusive scan over up to 1024 block totals.
__global__ void scan_tops_kernel(unsigned* __restrict__ bsums, int nb) {
    __shared__ unsigned tmp[1024];
    const int t = threadIdx.x;
    unsigned v = (t < nb) ? bsums[t] : 0u;
    tmp[t] = v;
    __syncthreads();
    #pragma unroll
    for (int d = 1; d < 1024; d <<= 1) {
        unsigned add = (t >= d) ? tmp[t - d] : 0u;
        __syncthreads();
        tmp[t] += add;
        __syncthreads();
    }
    if (t < nb) bsums[t] = tmp[t] - v;     // exclusive
}

__global__ void scan_add_kernel(unsigned* __restrict__ offs,
                                const unsigned* __restrict__ bsums,
                                unsigned* __restrict__ cursor, int n) {
    int g = blockIdx.x * 256 + threadIdx.x;
    if (g < n) {
        unsigned o = offs[g] + bsums[blockIdx.x];
        offs[g]   = o;
        cursor[g] = o;
    }
}

__global__ void scatter_kernel(const int* __restrict__ c, const int* __restrict__ s,
                               unsigned* __restrict__ cursor,
                               unsigned* __restrict__ order, int n) {
    int e = blockIdx.x * 256 + threadIdx.x;
    if (e < n) {
        unsigned pos = atomicAdd(&cursor[c[e] * 4 + s[e]], 1u);
        order[pos] = (unsigned)e;
    }
}

// Phase 2: one block per 8 centers. Stream sorted edges, WMMA radial GEMM,
// scatter outer products into an LDS accumulator (ds_add_f32), one coalesced
// store pass per block. d_out written exactly once -> no zero pass needed.
__global__ __launch_bounds__(256)
void sph_phase2_kernel(const float* __restrict__ vectors,
                       const float* __restrict__ rw,
                       const int*   __restrict__ center_idx,
                       const int*   __restrict__ species_idx,
                       const unsigned* __restrict__ offs,
                       const unsigned* __restrict__ order,
                       float* __restrict__ out,
                       int n_edges, int centers) {
    __shared__ __align__(16) float scratch[8][32][32]; // per-wave WMMA buffers
    __shared__ __align__(16) float acc[8 * 512];       // 8 centers x 16 x 32

    const int tid = threadIdx.x, wave = tid >> 5, lane = tid & 31;
    const int c0 = blockIdx.x * 8;

    #pragma unroll
    for (int k = 0; k < 16; ++k) acc[tid + k * 256] = 0.0f;

    v2f b[2][3];
    load_B(rw, lane, b);

    const int estart  = (int)offs[(size_t)c0 * 4];
    const int bin_end = (c0 + 8) * 4;
    const int eend    = (bin_end >= centers * 4) ? n_edges : (int)offs[bin_end];
    __syncthreads();

    for (int base = estart; base < eend; base += 256) {
        const int  i     = base + tid;
        const bool valid = (i < eend);
        int e = 0;
        float vx = 1.f, vy = 0.f, vz = 0.f;
        if (valid) {
            e  = (int)order[i];
            vx = vectors[3 * e + 0];
            vy = vectors[3 * e + 1];
            vz = vectors[3 * e + 2];
        }
        float phi[12], sh[16];
        edge_basis(vx, vy, vz, phi, sh);

        float rad[32];
        wmma_rad(scratch[wave], lane, b, phi, rad);   // full-EXEC, uniform loop

        if (valid) {
            const int c = center_idx[e], s = species_idx[e];
            float* p0 = &acc[(c - c0) * 512 + s * 8];
            #pragma unroll
            for (int j = 0; j < 16; ++j) {
                const int l = (j == 0) ? 0 : (j < 4) ? 1 : (j < 9) ? 2 : 3;
                const float shj = sh[j];
                float* p = p0 + j * 32;
                #pragma unroll
                for (int n = 0; n < 8; ++n)
                    atomicAdd(p + n, shj * rad[l * 8 + n]);  // ds_add_f32
            }
        }
    }
    __syncthreads();

    float4*       o4 = (float4*)(out + (size_t)c0 * 512);
    const float4* a4 = (const float4*)acc;
    #pragma unroll
    for (int k = 0; k < 4; ++k) o4[tid + k * 256] = a4[tid + k * 256];
}

// ---------------------------------------------------------------------------
// Fallback path (insufficient workspace): direct global-atomic scatter.
// ---------------------------------------------------------------------------
__global__ void zero_out_kernel(float4* __restrict__ out, int n4) {
    int i = blockIdx.x * blockDim.x + threadIdx.x;
    if (i < n4) out[i] = make_float4(0.f, 0.f, 0.f, 0.f);
}

__global__ __launch_bounds__(256)
void sph_direct_kernel(const float* __restrict__ vectors,
                       const float* __restrict__ rw,
                       const int*   __restrict__ center_idx,
                       const int*   __restrict__ species_idx,
                       float* __restrict__ out, int n_edges) {
    __shared__ __align__(16) float scratch[8][32][32];
    const int tid = threadIdx.x, wave = tid >> 5, lane = tid & 31;
    const int e = blockIdx.x * 256 + tid;
    const bool valid = (e < n_edges);

    float vx = 1.f, vy = 0.f, vz = 0.f;
    if (valid) {
        vx = vectors[3 * e + 0];
        vy = vectors[3 * e + 1];
        vz = vectors[3 * e + 2];
    }
    float phi[12], sh[16];
    edge_basis(vx, vy, vz, phi, sh);

    v2f b[2][3];
    load_B(rw, lane, b);

    float rad[32];
    wmma_rad(scratch[wave], lane, b, phi, rad);

    if (valid) {
        const int c = center_idx[e], s = species_idx[e];
        float* outp = out + (size_t)c * 512 + (size_t)s * 8;
        #pragma unroll
        for (int j = 0; j < 16; ++j) {
            const int l = (j == 0) ? 0 : (j < 4) ? 1 : (j < 9) ? 2 : 3;
            const float shj = sh[j];
            float* p = outp + j * 32;
            #pragma unroll
            for (int n = 0; n < 8; ++n)
                atomicAdd(p + n, shj * rad[l * 8 + n]);
        }
    }
}

// ---------------------------------------------------------------------------
extern "C" void kernel_launch(void* const* d_in, const int* in_sizes, int n_in,
                              void* d_out, int out_size, void* d_ws, size_t ws_size,
                              hipStream_t stream) {
    const float* vectors = (const float*)d_in[0];
    const float* rw      = (const float*)d_in[1];
    const int*   cidx    = (const int*)d_in[2];
    const int*   sidx    = (const int*)d_in[3];
    float*       out     = (float*)d_out;

    const int n_edges = in_sizes[2];
    const int centers = out_size / 512;      // out = centers * 16 * 32
    const int nbins   = centers * 4;
    const int nscanb  = (nbins + 255) / 256;

    // Workspace layout (u32): count | offs | cursor | bsums(1024) | order
    unsigned* ws_u   = (unsigned*)d_ws;
    unsigned* count  = ws_u;
    unsigned* offs   = ws_u + (size_t)nbins;
    unsigned* cursor = ws_u + (size_t)2 * nbins;
    unsigned* bsums  = ws_u + (size_t)3 * nbins;
    unsigned* order  = ws_u + (size_t)3 * nbins + 1024;
    const size_t need = ((size_t)3 * nbins + 1024 + (size_t)n_edges) * sizeof(unsigned);

    const int eblocks = (n_edges + 255) / 256;

    if (ws_size >= need && nscanb <= 1024 && (centers % 8) == 0) {
        // ---- sorted path: counting sort by (center,species), LDS accumulate
        fill_u32_kernel<<<(nbins + 255) / 256, 256, 0, stream>>>(count, 0u, nbins);
        hist_kernel<<<eblocks, 256, 0, stream>>>(cidx, sidx, count, n_edges);
        scan_block_kernel<<<nscanb, 256, 0, stream>>>(count, offs, bsums, nbins);
        scan_tops_kernel<<<1, 1024, 0, stream>>>(bsums, nscanb);
        scan_add_kernel<<<nscanb, 256, 0, stream>>>(offs, bsums, cursor, nbins);
        scatter_kernel<<<eblocks, 256, 0, stream>>>(cidx, sidx, cursor, order, n_edges);
        sph_phase2_kernel<<<centers / 8, 256, 0, stream>>>(
            vectors, rw, cidx, sidx, offs, order, out, n_edges, centers);
    } else {
        // ---- fallback: direct global-atomic scatter (L2-resident accumulator)
        const int n4 = out_size / 4;
        zero_out_kernel<<<(n4 + 255) / 256, 256, 0, stream>>>((float4*)out, n4);
        sph_direct_kernel<<<eblocks, 256, 0, stream>>>(
            vectors, rw, cidx, sidx, out, n_edges);
    }
}